// LSTMModel_60078002536801
// MI455X (gfx1250) — compile-verified
//
#include <hip/hip_runtime.h>
#include <math.h>

typedef __attribute__((ext_vector_type(2))) float v2f;
typedef __attribute__((ext_vector_type(8))) float v8f;

__device__ __forceinline__ float sigmoidf_(float x) {
    return 1.0f / (1.0f + __expf(-x));
}

// ---------------------------------------------------------------------------
// Input-projection GEMM:  XG[r, n] = sum_k X[row(r), k] * Wih[n, k] + bih[n] + bhh[n]
// Output row r corresponds to (t = r / B, b = r % B)  -> XG stored time-major [t][b][g].
// B_MAJOR_IN: layer-0 input x is [b][t][k]; later layers' hs are already [t][b][k].
// W is staged in LDS pair-interleaved along K (Wp[k/2][n][2], stride S2=2G+32)
// so each WMMA B-fragment is one aligned ds_load_b64.
// ---------------------------------------------------------------------------
template<int K, int G, bool B_MAJOR_IN>
__global__ __launch_bounds__(256) void xg_gemm_kernel(
    const float* __restrict__ X, const float* __restrict__ Wih,
    const float* __restrict__ bih, const float* __restrict__ bhh,
    float* __restrict__ XG, int Bsz, int T)
{
    constexpr int MB = 128;        // rows per workgroup
    constexpr int SA = K + 4;      // A row stride (even; lane*SA spans distinct banks)
    constexpr int S2 = 2 * G + 32; // Wp row stride (S2 mod 64 == 32 -> half-waves disjoint)
    extern __shared__ float lds[];
    float* Alds = lds;                   // [MB][SA]
    float* Wp   = lds + MB * SA;         // [K/2][S2]; Wp[kp][2n+r] = Wih[n][2kp+r]

    const int tid  = threadIdx.x;
    const int wave = tid >> 5;
    const int lane = tid & 31;
    const int half = lane >> 4;
    const int l16  = lane & 15;
    const int r0   = blockIdx.x * MB;

    // Stage W pair-interleaved (v2f read, v2f write).
    {
        const v2f* W2 = (const v2f*)Wih;
        for (int idx = tid; idx < G * (K / 2); idx += 256) {
            int n = idx / (K / 2), kp = idx - n * (K / 2);
            *(v2f*)&Wp[kp * S2 + 2 * n] = W2[idx];
        }
    }
    // Stage A block (v2f along k).
    for (int idx = tid; idx < MB * (K / 2); idx += 256) {
        int i = idx / (K / 2), kp = idx - i * (K / 2);
        int r = r0 + i;
        long srcRow;
        if (B_MAJOR_IN) {
            int t = r / Bsz; int b = r - t * Bsz;
            srcRow = ((long)b * T + t) * K;
        } else {
            srcRow = (long)r * K;
        }
        *(v2f*)&Alds[i * SA + 2 * kp] = *(const v2f*)&X[srcRow + 2 * kp];
    }
    __syncthreads();

    constexpr int NT = G / 16;
    for (int tt = wave; tt < 8 * NT; tt += 8) {
        int mt = tt & 7;
        int nt = tt >> 3;
        int n0 = nt * 16;
        v8f acc = {0.f, 0.f, 0.f, 0.f, 0.f, 0.f, 0.f, 0.f};
        const float* arow = &Alds[(mt * 16 + l16) * SA + 2 * half];
        const float* wb   = &Wp[half * S2 + (n0 + l16) * 2];
#pragma unroll
        for (int kk = 0; kk < K; kk += 4) {
            v2f a = *(const v2f*)(arow + kk);              // ds_load_b64
            v2f b = *(const v2f*)(wb + (kk >> 1) * S2);    // ds_load_b64
            acc = __builtin_amdgcn_wmma_f32_16x16x4_f32(
                false, a, false, b, (short)0, acc, false, false);
        }
        float bias = bih[n0 + l16] + bhh[n0 + l16];
#pragma unroll
        for (int j = 0; j < 8; ++j) {
            int m = mt * 16 + j + 8 * half;
            XG[(long)(r0 + m) * G + n0 + l16] = acc[j] + bias;
        }
    }
}

// ---------------------------------------------------------------------------
// LSTM recurrence. One workgroup owns 16 batch rows; W_hh.T lives entirely in
// LDS pair-interleaved (layer 0: 256 KB -> uses CDNA5's 320 KB WGP LDS).
// Wave w owns hidden columns [16w, 16w+16) => gate tiles {w, w+H/16, w+2H/16,
// w+3H/16}, so i/f/g/o combine element-wise in registers. c stays in WMMA
// C-fragments; h round-trips through a small LDS tile each step (layout
// transpose). HS (optional) written time-major [t][b][h] for the next layer.
// ---------------------------------------------------------------------------
template<int H>
__global__ __launch_bounds__(256) void lstm_rec_kernel(
    const float* __restrict__ XG, const float* __restrict__ Whh,
    float* __restrict__ HS, float* __restrict__ HLAST, int Bsz, int T)
{
    constexpr int G  = 4 * H;
    constexpr int S2 = 2 * G + 32; // Wp row stride (== 2*(G+16); S2 mod 64 == 32)
    constexpr int SH = H + 4;
    constexpr int NW = H / 16;     // waves per workgroup
    constexpr int NTHREADS = NW * 32;
    extern __shared__ float lds[];
    float* Wp   = lds;                   // [H/2][S2]; Wp[kp][2n+r] = Whh[n][2kp+r]
    float* hlds = lds + (H / 2) * S2;    // [16][SH]

    const int tid  = threadIdx.x;
    const int wave = tid >> 5;
    const int lane = tid & 31;
    const int half = lane >> 4;
    const int l16  = lane & 15;
    const int b0   = blockIdx.x * 16;
    const int hid0 = wave * 16;

    {
        const v2f* W2 = (const v2f*)Whh;
        for (int idx = tid; idx < G * (H / 2); idx += NTHREADS) {
            int n = idx / (H / 2), kp = idx - n * (H / 2);
            *(v2f*)&Wp[kp * S2 + 2 * n] = W2[idx];
        }
    }
    for (int idx = tid; idx < 16 * SH; idx += NTHREADS) hlds[idx] = 0.f;
    __syncthreads();

    v8f c = {0.f, 0.f, 0.f, 0.f, 0.f, 0.f, 0.f, 0.f};
    v8f h = {0.f, 0.f, 0.f, 0.f, 0.f, 0.f, 0.f, 0.f};

    for (int t = 0; t < T; ++t) {
        v8f acc[4];
#pragma unroll
        for (int gi = 0; gi < 4; ++gi) {
            const float* xg = &XG[((long)t * Bsz + b0) * G + hid0 + gi * H + l16];
#pragma unroll
            for (int j = 0; j < 8; ++j) acc[gi][j] = xg[(long)(j + 8 * half) * G];
        }
        const float* arow = &hlds[l16 * SH + 2 * half];
        const float* wb   = &Wp[half * S2 + (hid0 + l16) * 2];
#pragma unroll 8
        for (int kk = 0; kk < H; kk += 4) {
            v2f a = *(const v2f*)(arow + kk);                        // ds_load_b64
#pragma unroll
            for (int gi = 0; gi < 4; ++gi) {
                v2f b = *(const v2f*)(wb + (kk >> 1) * S2 + gi * (2 * H)); // ds_load_b64
                acc[gi] = __builtin_amdgcn_wmma_f32_16x16x4_f32(
                    false, a, false, b, (short)0, acc[gi], false, false);
            }
        }
#pragma unroll
        for (int j = 0; j < 8; ++j) {
            float ig = sigmoidf_(acc[0][j]);
            float fg = sigmoidf_(acc[1][j]);
            float gg = tanhf(acc[2][j]);
            float og = sigmoidf_(acc[3][j]);
            float cn = fg * c[j] + ig * gg;
            c[j] = cn;
            h[j] = og * tanhf(cn);
        }
        __syncthreads();   // all waves finished reading hlds for this step
#pragma unroll
        for (int j = 0; j < 8; ++j) {
            int m = j + 8 * half;
            hlds[m * SH + hid0 + l16] = h[j];
            if (HS) HS[((long)t * Bsz + (b0 + m)) * H + hid0 + l16] = h[j];
        }
        __syncthreads();   // new h visible to all waves
    }
    if (HLAST) {
#pragma unroll
        for (int j = 0; j < 8; ++j) {
            int m = j + 8 * half;
            HLAST[(long)(b0 + m) * H + hid0 + l16] = h[j];
        }
    }
}

// ---------------------------------------------------------------------------
// Two tiny MLP heads on the final hidden state (1024 x 32). One thread / row.
// ---------------------------------------------------------------------------
__global__ __launch_bounds__(256) void heads_kernel(
    const float* __restrict__ HL,
    const float* __restrict__ rW1, const float* __restrict__ rb1,
    const float* __restrict__ rW2, const float* __restrict__ rb2,
    const float* __restrict__ cW1, const float* __restrict__ cb1,
    const float* __restrict__ cW2, const float* __restrict__ cb2,
    float* __restrict__ out, int Bsz)
{
    int b = blockIdx.x * blockDim.x + threadIdx.x;
    if (b >= Bsz) return;
    float h[32];
#pragma unroll
    for (int k = 0; k < 32; ++k) h[k] = HL[b * 32 + k];
    float sreg = rb2[0];
    float sclf = cb2[0];
    for (int j = 0; j < 64; ++j) {
        float ar = rb1[j], ac = cb1[j];
#pragma unroll
        for (int k = 0; k < 32; ++k) {
            ar = fmaf(h[k], rW1[j * 32 + k], ar);
            ac = fmaf(h[k], cW1[j * 32 + k], ac);
        }
        ar = fmaxf(ar, 0.f);
        ac = fmaxf(ac, 0.f);
        sreg = fmaf(ar, rW2[j], sreg);
        sclf = fmaf(ac, cW2[j], sclf);
    }
    out[b]       = tanhf(sreg);
    out[Bsz + b] = sigmoidf_(sclf);
}

// ---------------------------------------------------------------------------
extern "C" void kernel_launch(void* const* d_in, const int* in_sizes, int n_in,
                              void* d_out, int out_size, void* d_ws, size_t ws_size,
                              hipStream_t stream) {
    constexpr int B = 1024, T = 256, F = 64;
    const float* x    = (const float*)d_in[0];
    // d_in[1] = horizon (unused by the reference computation)
    const float* Wih0 = (const float*)d_in[2];
    const float* Whh0 = (const float*)d_in[3];
    const float* bih0 = (const float*)d_in[4];
    const float* bhh0 = (const float*)d_in[5];
    const float* Wih1 = (const float*)d_in[6];
    const float* Whh1 = (const float*)d_in[7];
    const float* bih1 = (const float*)d_in[8];
    const float* bhh1 = (const float*)d_in[9];
    const float* Wih2 = (const float*)d_in[10];
    const float* Whh2 = (const float*)d_in[11];
    const float* bih2 = (const float*)d_in[12];
    const float* bhh2 = (const float*)d_in[13];
    const float* rW1 = (const float*)d_in[14];
    const float* rb1 = (const float*)d_in[15];
    const float* rW2 = (const float*)d_in[16];
    const float* rb2 = (const float*)d_in[17];
    const float* cW1 = (const float*)d_in[18];
    const float* cb1 = (const float*)d_in[19];
    const float* cW2 = (const float*)d_in[20];
    const float* cb2 = (const float*)d_in[21];

    // Workspace layout (floats): xg buffer reused across layers.
    float* ws    = (float*)d_ws;
    float* xg    = ws;                                 // up to B*T*512 = 134,217,728 floats
    float* hs0   = ws   + (size_t)B * T * 512;         // B*T*128
    float* hs1   = hs0  + (size_t)B * T * 128;         // B*T*64
    float* hlast = hs1  + (size_t)B * T * 64;          // B*32

    const int M = B * T;
    const int gemmBlocks = M / 128;

    // Dynamic-LDS sizes (bytes). (Pair-interleave keeps totals identical.)
    constexpr int SH_G0 = (128 * (64 + 4)  + 32 * (2 * 512 + 32)) * 4;  // 169,984
    constexpr int SH_G1 = (128 * (128 + 4) + 64 * (2 * 256 + 32)) * 4;  // 206,848
    constexpr int SH_G2 = (128 * (64 + 4)  + 32 * (2 * 128 + 32)) * 4;  //  71,680
    constexpr int SH_R0 = (64 * (2 * 512 + 32) + 16 * (128 + 4)) * 4;   // 278,784  (W_hh resident)
    constexpr int SH_R1 = (32 * (2 * 256 + 32) + 16 * (64 + 4))  * 4;   //  73,984
    constexpr int SH_R2 = (16 * (2 * 128 + 32) + 16 * (32 + 4))  * 4;   //  20,736

    hipFuncSetAttribute((const void*)&xg_gemm_kernel<64, 512, true>,
                        hipFuncAttributeMaxDynamicSharedMemorySize, SH_G0);
    hipFuncSetAttribute((const void*)&xg_gemm_kernel<128, 256, false>,
                        hipFuncAttributeMaxDynamicSharedMemorySize, SH_G1);
    hipFuncSetAttribute((const void*)&xg_gemm_kernel<64, 128, false>,
                        hipFuncAttributeMaxDynamicSharedMemorySize, SH_G2);
    hipFuncSetAttribute((const void*)&lstm_rec_kernel<128>,
                        hipFuncAttributeMaxDynamicSharedMemorySize, SH_R0);
    hipFuncSetAttribute((const void*)&lstm_rec_kernel<64>,
                        hipFuncAttributeMaxDynamicSharedMemorySize, SH_R1);
    hipFuncSetAttribute((const void*)&lstm_rec_kernel<32>,
                        hipFuncAttributeMaxDynamicSharedMemorySize, SH_R2);

    // Layer 0: F=64 -> H=128
    xg_gemm_kernel<64, 512, true><<<gemmBlocks, 256, SH_G0, stream>>>(
        x, Wih0, bih0, bhh0, xg, B, T);
    lstm_rec_kernel<128><<<B / 16, 256, SH_R0, stream>>>(
        xg, Whh0, hs0, nullptr, B, T);

    // Layer 1: 128 -> 64
    xg_gemm_kernel<128, 256, false><<<gemmBlocks, 256, SH_G1, stream>>>(
        hs0, Wih1, bih1, bhh1, xg, B, T);
    lstm_rec_kernel<64><<<B / 16, 128, SH_R1, stream>>>(
        xg, Whh1, hs1, nullptr, B, T);

    // Layer 2: 64 -> 32 (only final hidden state needed)
    xg_gemm_kernel<64, 128, false><<<gemmBlocks, 256, SH_G2, stream>>>(
        hs1, Wih2, bih2, bhh2, xg, B, T);
    lstm_rec_kernel<32><<<B / 16, 64, SH_R2, stream>>>(
        xg, Whh2, nullptr, hlast, B, T);

    // Heads -> d_out: [reg(1024), clf(1024)]
    heads_kernel<<<(B + 255) / 256, 256, 0, stream>>>(
        hlast, rW1, rb1, rW2, rb2, cW1, cb1, cW2, cb2, (float*)d_out, B);
}